// ConcatenateSumOperation2_48773648613702
// MI455X (gfx1250) — compile-verified
//
#include <hip/hip_runtime.h>

// Ragged per-tensor sum over seq dim fused with feature concat:
//   x_i : [B=512, L_i, D=128] f32, L_i = (i+1)*64, i = 0..7
//   out : [512, 1024] f32, out[b, i*128 + d] = sum_l x_i[b, l, d]
//
// Memory-bound: ~604 MB streamed reads vs 2 MB writes. Strategy:
//  - 1 block (256 thr = 8 wave32) per (i, b); input slice is contiguous.
//  - b128 non-temporal coalesced loads, 4 independent accumulators/thread.
//  - 8x128 partials in LDS; column-sum finished with V_WMMA_F32_16X16X4_F32
//    (A == ones => C[m,n] = sum_k B[k,n]), two chained WMMAs per wave.

typedef __attribute__((ext_vector_type(4))) float fx4;
typedef __attribute__((ext_vector_type(2))) float v2f;
typedef __attribute__((ext_vector_type(8))) float v8f;

__global__ __launch_bounds__(256) void ragged_sum_concat_kernel(
    const float* __restrict__ x0, const float* __restrict__ x1,
    const float* __restrict__ x2, const float* __restrict__ x3,
    const float* __restrict__ x4, const float* __restrict__ x5,
    const float* __restrict__ x6, const float* __restrict__ x7,
    float* __restrict__ out)
{
    constexpr int D = 128;
    __shared__ float lds[8 * D];   // 8 l-stripes x 128 feature columns

    const int i = (int)(blockIdx.x & 7);   // tensor index (low bits: balance mix)
    const int b = (int)(blockIdx.x >> 3);  // batch row

    const int L = (i + 1) * 64;            // ragged length

    const float* base;
    switch (i) {                           // uniform per block -> scalar branches
        case 0:  base = x0; break;
        case 1:  base = x1; break;
        case 2:  base = x2; break;
        case 3:  base = x3; break;
        case 4:  base = x4; break;
        case 5:  base = x5; break;
        case 6:  base = x6; break;
        default: base = x7; break;
    }
    base += (size_t)b * (size_t)L * D;

    const int t  = (int)threadIdx.x;
    const int d4 = t & 31;   // float4 column 0..31  (wave covers 512 contiguous B)
    const int s  = t >> 5;   // l-stripe 0..7 == wave id (wave32, packed order)

    // ---- streaming accumulation: rows l = s, s+8, s+16, ... --------------
    fx4 a0 = {0.f, 0.f, 0.f, 0.f};
    fx4 a1 = {0.f, 0.f, 0.f, 0.f};
    fx4 a2 = {0.f, 0.f, 0.f, 0.f};
    fx4 a3 = {0.f, 0.f, 0.f, 0.f};
    const fx4* col = (const fx4*)base + d4;   // row stride = D/4 = 32 fx4
    for (int l = s; l < L; l += 32) {         // L % 32 == 0 for all i
        a0 += __builtin_nontemporal_load(&col[(size_t)(l     ) * 32]);
        a1 += __builtin_nontemporal_load(&col[(size_t)(l +  8) * 32]);
        a2 += __builtin_nontemporal_load(&col[(size_t)(l + 16) * 32]);
        a3 += __builtin_nontemporal_load(&col[(size_t)(l + 24) * 32]);
    }
    fx4 acc = (a0 + a1) + (a2 + a3);

    ((fx4*)lds)[s * 32 + d4] = acc;
    __syncthreads();

    // ---- 8x128 -> 128 column sums via WMMA (A = ones) --------------------
    // Each wave reduces a 16-column chunk n0 = s*16.
    // B 4x16 f32 tile: N = lane%16; the K <-> (slot, lane-half) assignment is
    // irrelevant under A==1 (column sums are row-permutation invariant).
    const int lane = t & 31;
    const int n    = lane & 15;
    const int half = lane >> 4;
    const int n0   = s * 16;

    v2f Bm0, Bm1;
    Bm0[0] = lds[(0 + half) * D + n0 + n];   // rows 0,1
    Bm0[1] = lds[(2 + half) * D + n0 + n];   // rows 2,3
    Bm1[0] = lds[(4 + half) * D + n0 + n];   // rows 4,5
    Bm1[1] = lds[(6 + half) * D + n0 + n];   // rows 6,7

    v2f A1; A1[0] = 1.0f; A1[1] = 1.0f;
    v8f c = {};
    // (neg_a, A, neg_b, B, c_mod, C, reuse_a, reuse_b)
    c = __builtin_amdgcn_wmma_f32_16x16x4_f32(false, A1, false, Bm0,
                                              (short)0, c, false, false);
    c = __builtin_amdgcn_wmma_f32_16x16x4_f32(false, A1, false, Bm1,
                                              (short)0, c, false, false);

    // C/D layout: VGPR0, lanes 0..15 hold (M=0, N=lane) = column sums.
    if (lane < 16) {
        out[(size_t)b * (8 * D) + (size_t)i * D + n0 + n] = c[0];
    }
}

extern "C" void kernel_launch(void* const* d_in, const int* in_sizes, int n_in,
                              void* d_out, int out_size, void* d_ws, size_t ws_size,
                              hipStream_t stream) {
    (void)in_sizes; (void)n_in; (void)out_size; (void)d_ws; (void)ws_size;
    const float* x0 = (const float*)d_in[0];
    const float* x1 = (const float*)d_in[1];
    const float* x2 = (const float*)d_in[2];
    const float* x3 = (const float*)d_in[3];
    const float* x4 = (const float*)d_in[4];
    const float* x5 = (const float*)d_in[5];
    const float* x6 = (const float*)d_in[6];
    const float* x7 = (const float*)d_in[7];
    float* out = (float*)d_out;

    dim3 grid(8 * 512);   // one block per (tensor, batch-row)
    dim3 block(256);      // 8 wave32
    ragged_sum_concat_kernel<<<grid, block, 0, stream>>>(
        x0, x1, x2, x3, x4, x5, x6, x7, out);
}